// PostprocessWithSampling_33457795235912
// MI455X (gfx1250) — compile-verified
//
#include <hip/hip_runtime.h>

// Problem constants (match reference).
#define BATCH   128
#define VOCAB   128000
#define MAX_SEQ 4096

// Argmax decomposition: 4 blocks per batch row, 256 threads per block.
#define BLOCKS_PER_ROW 4
#define NPART   (BATCH * BLOCKS_PER_ROW)      // 512 partials
#define CHUNK   (VOCAB / BLOCKS_PER_ROW)      // 32000 floats per block
#define CHUNK4  (CHUNK / 4)                   // 8000 float4 per block
#define ATHREADS 256

// Native clang vector types (required for __builtin_nontemporal_load/store).
typedef float vfloat4 __attribute__((ext_vector_type(4)));
typedef int   vint4   __attribute__((ext_vector_type(4)));

// Output layout (flat fp32, reference tuple order):
//   [0, 128)                      tokens
//   [128, 256)                    last_token_index (updated)
//   [256, 256+524288)             attention_mask  (updated)
//   [524544, 1048832)             generated_tokens (updated)
//   [1048832]                     generated_index  (updated)
#define OUT_TOK   0
#define OUT_LTI   (BATCH)
#define OUT_MASK  (2 * BATCH)
#define OUT_GTOK  (2 * BATCH + BATCH * MAX_SEQ)
#define OUT_GIDX  (2 * BATCH + 2 * BATCH * MAX_SEQ)

// ---------------------------------------------------------------------------
// Kernel 1: streaming per-chunk argmax of logits. One block owns a contiguous
// 32000-float chunk of one row. Non-temporal b128 loads (read-once data) +
// global_prefetch_b8 for the next strided tile. Partial (val, idx) -> d_ws.
// ---------------------------------------------------------------------------
__global__ __launch_bounds__(ATHREADS) void argmax_partial_kernel(
    const float* __restrict__ logits,
    float* __restrict__ pval, int* __restrict__ pidx) {
  const int blk = blockIdx.x;                 // 0..511
  const int b   = blk / BLOCKS_PER_ROW;       // batch row
  const int j   = blk % BLOCKS_PER_ROW;       // chunk within row
  const int tid = threadIdx.x;

  const vfloat4* __restrict__ src =
      (const vfloat4*)(logits + (size_t)b * VOCAB + (size_t)j * CHUNK);

  float bestv = -__builtin_huge_valf();
  int   besti = 0;

  // Strided streaming loop; indices ascend within a thread, so strict '>'
  // preserves first-occurrence semantics of jnp.argmax.
  for (int i = tid; i < CHUNK4; i += ATHREADS) {
    // gfx1250: global_prefetch_b8 of the next tile (speculative OK at tail).
    __builtin_prefetch((const void*)(src + i + ATHREADS), 0, 0);
    vfloat4 v = __builtin_nontemporal_load(src + i);   // TH=NT b128 load
    const int gi = j * CHUNK + i * 4;
    if (v.x > bestv) { bestv = v.x; besti = gi;     }
    if (v.y > bestv) { bestv = v.y; besti = gi + 1; }
    if (v.z > bestv) { bestv = v.z; besti = gi + 2; }
    if (v.w > bestv) { bestv = v.w; besti = gi + 3; }
  }

  __shared__ float sv[ATHREADS];
  __shared__ int   si[ATHREADS];
  sv[tid] = bestv;
  si[tid] = besti;
  __syncthreads();

  // Tree reduction with first-occurrence tie-break (smaller index wins ties).
  for (int off = ATHREADS / 2; off > 0; off >>= 1) {
    if (tid < off) {
      const float ov = sv[tid + off];
      const int   oi = si[tid + off];
      if (ov > sv[tid] || (ov == sv[tid] && oi < si[tid])) {
        sv[tid] = ov;
        si[tid] = oi;
      }
    }
    __syncthreads();
  }

  if (tid == 0) {
    pval[blk] = sv[0];
    pidx[blk] = si[0];
  }
}

// ---------------------------------------------------------------------------
// Kernel 2: finalize. 128 threads: fold 4 partials per row -> token, emit
// tokens + updated last_token_index; thread 0 emits updated generated_index.
// ---------------------------------------------------------------------------
__global__ void finalize_kernel(const int* __restrict__ last_token_index,
                                const int* __restrict__ generated_index,
                                const float* __restrict__ pval,
                                const int* __restrict__ pidx,
                                int* __restrict__ wtok,
                                float* __restrict__ out) {
  const int b = threadIdx.x;
  if (b < BATCH) {
    float bv = pval[b * BLOCKS_PER_ROW];
    int   bi = pidx[b * BLOCKS_PER_ROW];
#pragma unroll
    for (int j = 1; j < BLOCKS_PER_ROW; ++j) {
      const float v = pval[b * BLOCKS_PER_ROW + j];
      const int   i = pidx[b * BLOCKS_PER_ROW + j];
      if (v > bv || (v == bv && i < bi)) { bv = v; bi = i; }
    }
    out[OUT_TOK + b] = (float)bi;   // greedy token
    wtok[b] = bi;                   // int copy for the scatter kernel

    int nl = last_token_index[b] + 1;
    if (nl > MAX_SEQ - 1) nl = MAX_SEQ - 1;
    out[OUT_LTI + b] = (float)nl;
  }
  if (b == 0) {
    int ng = generated_index[0] + 1;
    if (ng > MAX_SEQ - 1) ng = MAX_SEQ - 1;
    out[OUT_GIDX] = (float)ng;
  }
}

// ---------------------------------------------------------------------------
// Kernel 3: copy attention_mask / generated_tokens to output (as fp32) with
// the two scatters fused in. b128 loads, non-temporal b128 stores (4 elems
// per thread). attention_mask scatter uses the POST-increment clamped index;
// the generated_tokens scatter uses the PRE-increment generated_index
// (reference order of operations).
// ---------------------------------------------------------------------------
__global__ void scatter_copy_kernel(const int* __restrict__ last_token_index,
                                    const int* __restrict__ generated_index,
                                    const int* __restrict__ in_mask,
                                    const int* __restrict__ in_tok,
                                    const int* __restrict__ wtok,
                                    float* __restrict__ out) {
  const int i4 = blockIdx.x * blockDim.x + threadIdx.x;
  const int n4 = (BATCH * MAX_SEQ) / 4;
  if (i4 >= n4) return;

  const int base = i4 * 4;
  const int b = base >> 12;        // / MAX_SEQ
  const int s = base & (MAX_SEQ - 1);

  const vint4 mv = __builtin_nontemporal_load((const vint4*)in_mask + i4);
  const vint4 tv = __builtin_nontemporal_load((const vint4*)in_tok + i4);

  int clamped = last_token_index[b] + 1;
  if (clamped > MAX_SEQ - 1) clamped = MAX_SEQ - 1;
  const int col = generated_index[0];          // pre-increment column
  const int tok = wtok[b];

  vfloat4 mo;
  mo.x = (float)mv.x; mo.y = (float)mv.y; mo.z = (float)mv.z; mo.w = (float)mv.w;
  vfloat4 to;
  to.x = (float)tv.x; to.y = (float)tv.y; to.z = (float)tv.z; to.w = (float)tv.w;

  if (clamped >= s && clamped < s + 4) {
    const int k = clamped - s;
    if (k == 0) mo.x = 1.0f; else if (k == 1) mo.y = 1.0f;
    else if (k == 2) mo.z = 1.0f; else mo.w = 1.0f;
  }
  if (col >= s && col < s + 4) {
    const int k = col - s;
    const float tf = (float)tok;
    if (k == 0) to.x = tf; else if (k == 1) to.y = tf;
    else if (k == 2) to.z = tf; else to.w = tf;
  }

  __builtin_nontemporal_store(mo, (vfloat4*)(out + OUT_MASK) + i4);
  __builtin_nontemporal_store(to, (vfloat4*)(out + OUT_GTOK) + i4);
}

// ---------------------------------------------------------------------------
extern "C" void kernel_launch(void* const* d_in, const int* in_sizes, int n_in,
                              void* d_out, int out_size, void* d_ws, size_t ws_size,
                              hipStream_t stream) {
  const float* logits = (const float*)d_in[0];
  const int* lti      = (const int*)d_in[1];
  const int* mask     = (const int*)d_in[2];
  const int* gtok     = (const int*)d_in[3];
  const int* gidx     = (const int*)d_in[4];
  float* out          = (float*)d_out;

  // Workspace layout: 512 partial values, 512 partial indices, 128 int tokens.
  float* pval = (float*)d_ws;
  int*   pidx = (int*)d_ws + NPART;
  int*   wtok = (int*)d_ws + 2 * NPART;

  argmax_partial_kernel<<<NPART, ATHREADS, 0, stream>>>(logits, pval, pidx);
  finalize_kernel<<<1, BATCH, 0, stream>>>(lti, gidx, pval, pidx, wtok, out);

  const int n4 = (BATCH * MAX_SEQ) / 4;        // 131072 vector elements
  scatter_copy_kernel<<<(n4 + 255) / 256, 256, 0, stream>>>(lti, gidx, mask,
                                                            gtok, wtok, out);
}